// ProjectionLoss_72275709657663
// MI455X (gfx1250) — compile-verified
//
#include <hip/hip_runtime.h>

// ProjectionLoss for MI455X (gfx1250, wave32).
// Core dot-products done with V_WMMA_F32_16X16X4_F32 (D = A[16x4] * B[4x16] + C).
//   A = projection rows (K*[R|t] or K*cam), B = homogeneous keypoints.
// One wave32 per batch element; 16 WMMAs per batch element (fully unrolled).

typedef __attribute__((ext_vector_type(2))) float v2f;
typedef __attribute__((ext_vector_type(8))) float v8f;

#define B_TOTAL 16384
#define V_VIEWS 8
#define J_KPS   64
#define WAVES_PER_BLOCK 8
#define THREADS (WAVES_PER_BLOCK * 32)

// fast reciprocal: v_rcp_f32 + one Newton step (error ~2^-28, well below loss tolerance)
__device__ __forceinline__ float fast_rcp(float z) {
    float r = __builtin_amdgcn_rcpf(z);
    return r * (2.0f - z * r);
}

__global__ __launch_bounds__(THREADS) void proj_loss_main(
    const float* __restrict__ kg,    // [B,J,3]  kps_world_gt
    const float* __restrict__ kp,    // [B,J,3]  kps_world_pred
    const float* __restrict__ Rg,    // [B,V,3,3] gt_R
    const float* __restrict__ tg,    // [B,V,3]   gt_t
    const float* __restrict__ Kin,   // [V,3,3]   intrinsics
    const float* __restrict__ cam,   // [B,V,3,4] cam_preds
    float* __restrict__ ws)          // [gridDim.x] per-block partial sums
{
    // Per-wave projection matrices P = K @ [R|t] (gt) and K @ cam (pred), [V][3][4]
    __shared__ float Pg[WAVES_PER_BLOCK][V_VIEWS][3][4];
    __shared__ float Pp[WAVES_PER_BLOCK][V_VIEWS][3][4];
    __shared__ float wsum[WAVES_PER_BLOCK];

    const int tid  = threadIdx.x;
    const int wave = tid >> 5;
    const int lane = tid & 31;
    const size_t b = (size_t)blockIdx.x * WAVES_PER_BLOCK + wave;

    // ---------------- Phase 1: build projection matrices into LDS ----------
    {
        const int v = lane >> 2;   // 0..7  view
        const int c = lane & 3;    // 0..3  column of the 3x4
        const float* Kv = Kin + v * 9;
        const float* Rb = Rg + (b * V_VIEWS + v) * 9;
        const float* tb = tg + (b * V_VIEWS + v) * 3;
        // column c of E = [R|t]
        float e0, e1, e2;
        if (c < 3) { e0 = Rb[0*3 + c]; e1 = Rb[1*3 + c]; e2 = Rb[2*3 + c]; }
        else       { e0 = tb[0];       e1 = tb[1];       e2 = tb[2];       }
        Pg[wave][v][0][c] = Kv[0]*e0 + Kv[1]*e1 + Kv[2]*e2;
        Pg[wave][v][1][c] = Kv[3]*e0 + Kv[4]*e1 + Kv[5]*e2;
        Pg[wave][v][2][c] = Kv[6]*e0 + Kv[7]*e1 + Kv[8]*e2;
        const float* Cb = cam + (b * V_VIEWS + v) * 12;
        float f0 = Cb[0*4 + c], f1 = Cb[1*4 + c], f2 = Cb[2*4 + c];
        Pp[wave][v][0][c] = Kv[0]*f0 + Kv[1]*f1 + Kv[2]*f2;
        Pp[wave][v][1][c] = Kv[3]*f0 + Kv[4]*f1 + Kv[5]*f2;
        Pp[wave][v][2][c] = Kv[6]*f0 + Kv[7]*f1 + Kv[8]*f2;
    }
    __syncthreads();

    // ---------------- Phase 2: WMMA projections + smooth-MSE ---------------
    // A-matrix 16x4 f32 layout: lane L holds M = L&15; K = {0,1} for lanes 0-15,
    // K = {2,3} for lanes 16-31 (VGPR0 = first K, VGPR1 = second K).
    const int hi   = lane >> 4;   // lane half
    const int mrow = lane & 15;   // A: M row index; B/D: N column index
    const int ks   = hi * 2;      // first K slot held by this lane

    // Row mapping per view-tile t (views 4t..4t+3):
    //   M in [0,5]  -> view 4t + M/3,        row M%3
    //   M in [8,13] -> view 4t+2 + (M-8)/3,  row (M-8)%3
    //   M in {6,7,14,15} -> zero row
    v2f Ag[2], Ap[2];
    {
        const int mlocal = mrow & 7;
        const bool valid = (mlocal < 6);
        const int vsub   = (mrow >> 3) * 2 + (valid ? mlocal / 3 : 0);
        const int r      = valid ? (mlocal % 3) : 0;
#pragma unroll
        for (int t = 0; t < 2; ++t) {
            const int v = 4 * t + vsub;
            float g0 = Pg[wave][v][r][ks], g1 = Pg[wave][v][r][ks + 1];
            float p0 = Pp[wave][v][r][ks], p1 = Pp[wave][v][r][ks + 1];
            Ag[t].x = valid ? g0 : 0.0f;  Ag[t].y = valid ? g1 : 0.0f;
            Ap[t].x = valid ? p0 : 0.0f;  Ap[t].y = valid ? p1 : 0.0f;
        }
    }

    const float THR  = 400.0f;
    const float TPOW = 219.712036f;  // 400^0.9
    float acc = 0.0f;

    // Base pointers for this lane's keypoint column; m-tiles use constant offsets.
    const float* Xgb = kg + (b * J_KPS + mrow) * 3;
    const float* Xpb = kp + (b * J_KPS + mrow) * 3;

#pragma unroll
    for (int m = 0; m < 4; ++m) {
        const float* Xg = Xgb + m * 16 * 3;
        const float* Xp = Xpb + m * 16 * 3;
        // B-matrix 4x16: lane holds column n = mrow; lanes 0-15 carry K={0,1},
        // lanes 16-31 carry K={2,3}. Homogeneous keypoint = (x,y,z,1).
        float gx = Xg[0], gy = Xg[1], gz = Xg[2];
        float px = Xp[0], py = Xp[1], pz = Xp[2];
        v2f Bg, Bp;
        Bg.x = hi ? gz : gx;  Bg.y = hi ? 1.0f : gy;
        Bp.x = hi ? pz : px;  Bp.y = hi ? 1.0f : py;

#pragma unroll
        for (int t = 0; t < 2; ++t) {
            v8f cg = {};
            v8f cp = {};
            cg = __builtin_amdgcn_wmma_f32_16x16x4_f32(
                     false, Ag[t], false, Bg, (short)0, cg, false, false);
            cp = __builtin_amdgcn_wmma_f32_16x16x4_f32(
                     false, Ap[t], false, Bp, (short)0, cp, false, false);
            // D: lane L, VGPR g -> M = g + 8*hi, N = L&15.
            // This lane owns keypoint j, views v0=4t+2*hi (g=0..2) and v0+1 (g=3..5).
#pragma unroll
            for (int s = 0; s < 2; ++s) {
                float xg = cg[3*s + 0], yg = cg[3*s + 1], zg = cg[3*s + 2];
                float xq = cp[3*s + 0], yq = cp[3*s + 1], zq = cp[3*s + 2];
                float izg = fast_rcp(zg);
                float izq = fast_rcp(zq);
                float dx = xg * izg - xq * izq;
                float dy = yg * izg - yq * izq;
                float d2x = dx * dx, d2y = dy * dy;
                // smooth threshold reweight (branchless, keeps EXEC full):
                // pow(d, 0.1) * 400^0.9 via hardware v_log_f32 / v_exp_f32
                float bx = __builtin_amdgcn_exp2f(0.1f * __builtin_amdgcn_logf(d2x)) * TPOW;
                float by = __builtin_amdgcn_exp2f(0.1f * __builtin_amdgcn_logf(d2y)) * TPOW;
                d2x = (d2x > THR) ? bx : d2x;
                d2y = (d2y > THR) ? by : d2y;
                acc += d2x + d2y;
            }
        }
    }

    // ---------------- deterministic reduction -----------------------------
#pragma unroll
    for (int off = 16; off > 0; off >>= 1)
        acc += __shfl_down(acc, off, 32);
    if (lane == 0) wsum[wave] = acc;
    __syncthreads();
    if (tid == 0) {
        float s = 0.0f;
#pragma unroll
        for (int w = 0; w < WAVES_PER_BLOCK; ++w) s += wsum[w];
        ws[blockIdx.x] = s;
    }
}

__global__ __launch_bounds__(256) void proj_loss_finish(
    const float* __restrict__ ws, int n, float* __restrict__ out)
{
    __shared__ float red[256];
    float s = 0.0f;
    for (int i = threadIdx.x; i < n; i += 256) s += ws[i];  // fixed order
    red[threadIdx.x] = s;
    __syncthreads();
#pragma unroll
    for (int k = 128; k > 0; k >>= 1) {
        if (threadIdx.x < k) red[threadIdx.x] += red[threadIdx.x + k];
        __syncthreads();
    }
    // per-batch: sum/dim (dim=2); then mean over B
    if (threadIdx.x == 0) out[0] = red[0] * (1.0f / (2.0f * (float)B_TOTAL));
}

extern "C" void kernel_launch(void* const* d_in, const int* in_sizes, int n_in,
                              void* d_out, int out_size, void* d_ws, size_t ws_size,
                              hipStream_t stream) {
    const float* kg  = (const float*)d_in[0];  // kps_world_gt
    const float* kp  = (const float*)d_in[1];  // kps_world_pred
    const float* Rg  = (const float*)d_in[2];  // gt_R
    const float* tg  = (const float*)d_in[3];  // gt_t
    const float* Kin = (const float*)d_in[4];  // K
    const float* cam = (const float*)d_in[5];  // cam_preds

    float* out = (float*)d_out;
    float* ws  = (float*)d_ws;   // needs 2048 floats = 8 KB scratch

    const int blocks = B_TOTAL / WAVES_PER_BLOCK;  // 2048, all waves full
    proj_loss_main<<<blocks, THREADS, 0, stream>>>(kg, kp, Rg, tg, Kin, cam, ws);
    proj_loss_finish<<<1, 256, 0, stream>>>(ws, blocks, out);
    (void)in_sizes; (void)n_in; (void)out_size; (void)ws_size;
}